// GATSubLayer_26998164423437
// MI455X (gfx1250) — compile-verified
//
#include <hip/hip_runtime.h>

// ---------------- constants (match reference) ----------------
#define D 128            // d_in == d_out == 128
#define TILE 16

typedef float v2f __attribute__((ext_vector_type(2)));
typedef float v8f __attribute__((ext_vector_type(8)));

// Monotone order-preserving float <-> uint encoding (for atomicMax over floats,
// including negatives). Key 0 is below every real key (even -inf -> 0x007FFFFF).
__device__ __forceinline__ unsigned f2ord(float f) {
    unsigned u = __float_as_uint(f);
    return (u & 0x80000000u) ? ~u : (u | 0x80000000u);
}
__device__ __forceinline__ float ord2f(unsigned k) {
    return __uint_as_float((k & 0x80000000u) ? (k & 0x7FFFFFFFu) : ~k);
}

// ---------------- 0. zero-init out / denom / mkey ----------------
__global__ void init_zero(float* __restrict__ out, float* __restrict__ denom,
                          unsigned* __restrict__ mkey, int n_out, int n_nodes) {
    int i = blockIdx.x * blockDim.x + threadIdx.x;
    int stride = gridDim.x * blockDim.x;
    for (int j = i; j < n_out; j += stride) out[j] = 0.0f;
    for (int j = i; j < n_nodes; j += stride) { denom[j] = 0.0f; mkey[j] = 0u; }
}

// ---------------- 1. z = h @ W via V_WMMA_F32_16X16X4_F32 ----------------
// One wave per 16x16 output tile. Block = 256 threads = 8 waves = 8 N-tiles
// (covers all 128 output columns for one 16-row stripe).
// A layout (32-bit, 16x4): lanes 0-15 row M=lane, VGPR0/1 = K,K+1; lanes 16-31
// same rows, K+2,K+3.  B layout (4x16): VGPR0 = row K (lanes 0-15, N=lane) /
// row K+2 (lanes 16-31); VGPR1 = rows K+1 / K+3.  C/D: VGPR r -> M=r (lo half)
// or M=r+8 (hi half), N = lane&15.
__global__ void gemm_wmma(const float* __restrict__ h, const float* __restrict__ W,
                          float* __restrict__ z, int n_nodes) {
    const int wave  = threadIdx.x >> 5;     // 0..7 -> N tile
    const int lane  = threadIdx.x & 31;
    const int hi    = lane >> 4;            // 0: lanes 0-15, 1: lanes 16-31
    const int l     = lane & 15;
    const int tileM = blockIdx.x * TILE;
    const int tileN = wave * TILE;

    // clamp keeps EXEC full for WMMA even on a ragged last tile
    const int row = (tileM + l < n_nodes) ? (tileM + l) : (n_nodes - 1);
    const float* __restrict__ arow = h + (size_t)row * D;

    v8f c = {};
#pragma unroll
    for (int k = 0; k < D; k += 4) {
        const int ka = k + 2 * hi;
        v2f a, b;
        a.x = arow[ka];
        a.y = arow[ka + 1];
        b.x = W[(size_t)ka * D + tileN + l];
        b.y = W[(size_t)(ka + 1) * D + tileN + l];
        c = __builtin_amdgcn_wmma_f32_16x16x4_f32(
            /*neg_a=*/false, a, /*neg_b=*/false, b,
            /*c_mod=*/(short)0, c, /*reuse_a=*/false, /*reuse_b=*/false);
    }
#pragma unroll
    for (int r = 0; r < 8; ++r) {
        const int m = tileM + r + 8 * hi;
        if (m < n_nodes) z[(size_t)m * D + tileN + l] = c[r];
    }
}

// ---------------- 2. per-node scores s_src = z.a_l, s_dst = z.a_r ----------------
__global__ void node_scores(const float* __restrict__ z, const float* __restrict__ attn,
                            float* __restrict__ s_src, float* __restrict__ s_dst,
                            int n_nodes) {
    const int wave = threadIdx.x >> 5;
    const int lane = threadIdx.x & 31;
    const int node = blockIdx.x * 8 + wave;
    if (node >= n_nodes) return;

    const float4 zv = ((const float4*)(z + (size_t)node * D))[lane];
    const float4 al = ((const float4*)attn)[lane];          // attn[0:128]
    const float4 ar = ((const float4*)(attn + D))[lane];    // attn[128:256]

    float ss = zv.x * al.x + zv.y * al.y + zv.z * al.z + zv.w * al.w;
    float sd = zv.x * ar.x + zv.y * ar.y + zv.z * ar.z + zv.w * ar.w;
#pragma unroll
    for (int off = 16; off > 0; off >>= 1) {
        ss += __shfl_xor(ss, off, 32);
        sd += __shfl_xor(sd, off, 32);
    }
    if (lane == 0) { s_src[node] = ss; s_dst[node] = sd; }
}

// ---------------- 3. edge logits + segment max ----------------
__global__ void edge_logits(const float* __restrict__ s_src, const float* __restrict__ s_dst,
                            const int* __restrict__ src, const int* __restrict__ dst,
                            float* __restrict__ ebuf, unsigned* __restrict__ mkey,
                            int n_edges) {
    const int i = blockIdx.x * blockDim.x + threadIdx.x;
    if (i >= n_edges) return;
    const float x = s_src[src[i]] + s_dst[dst[i]];
    const float e = (x > 0.0f) ? x : 0.01f * x;   // leaky_relu slope 0.01
    ebuf[i] = e;
    atomicMax(&mkey[dst[i]], f2ord(e));
}

// ---------------- 4. exp(e - max) + segment sum ----------------
__global__ void edge_exp(float* __restrict__ ebuf, const int* __restrict__ dst,
                         const unsigned* __restrict__ mkey, float* __restrict__ denom,
                         int n_edges) {
    const int i = blockIdx.x * blockDim.x + threadIdx.x;
    if (i >= n_edges) return;
    const int d = dst[i];
    const float ex = __expf(ebuf[i] - ord2f(mkey[d]));
    ebuf[i] = ex;                       // overwrite in place
    atomicAdd(&denom[d], ex);
}

// ---------------- 5. weighted scatter of source messages ----------------
// One wave per edge: 32 lanes x float4 = 128 floats of z[src].
__global__ void edge_scatter(const float* __restrict__ z, const float* __restrict__ exbuf,
                             const float* __restrict__ denom, const float* __restrict__ rel_emb,
                             const int* __restrict__ src, const int* __restrict__ dst,
                             const int* __restrict__ etype, float* __restrict__ out,
                             int n_edges) {
    const int wave = threadIdx.x >> 5;
    const int lane = threadIdx.x & 31;
    const int e = blockIdx.x * 8 + wave;
    if (e >= n_edges) return;

    const int s = src[e];
    const int d = dst[e];
    const int t = etype[e];
    const float rel = (t == 0) ? 0.0f : rel_emb[t];   // padding_idx = 0
    const float coef = rel * exbuf[e] / denom[d];

    const float4 zv = ((const float4*)(z + (size_t)s * D))[lane];
    float* op = out + (size_t)d * D + lane * 4;
    atomicAdd(op + 0, coef * zv.x);
    atomicAdd(op + 1, coef * zv.y);
    atomicAdd(op + 2, coef * zv.z);
    atomicAdd(op + 3, coef * zv.w);
}

// ---------------- launch ----------------
extern "C" void kernel_launch(void* const* d_in, const int* in_sizes, int n_in,
                              void* d_out, int out_size, void* d_ws, size_t ws_size,
                              hipStream_t stream) {
    const float* h    = (const float*)d_in[0];
    const float* W    = (const float*)d_in[1];
    const float* attn = (const float*)d_in[2];
    const float* rel  = (const float*)d_in[3];
    const int*   src  = (const int*)d_in[4];
    const int*   dst  = (const int*)d_in[5];
    const int*   ety  = (const int*)d_in[6];

    const int N = in_sizes[0] / D;      // 50000
    const int E = in_sizes[4];          // 800000

    float* out = (float*)d_out;

    // workspace carve-out (256B aligned)
    char* ws = (char*)d_ws;
    size_t off = 0;
    auto carve = [&](size_t bytes) -> void* {
        void* p = ws + off;
        off = (off + bytes + 255) & ~(size_t)255;
        return p;
    };
    float*    z     = (float*)carve((size_t)N * D * sizeof(float));
    float*    ssrc  = (float*)carve((size_t)N * sizeof(float));
    float*    sdst  = (float*)carve((size_t)N * sizeof(float));
    unsigned* mkey  = (unsigned*)carve((size_t)N * sizeof(unsigned));
    float*    denom = (float*)carve((size_t)N * sizeof(float));
    float*    ebuf  = (float*)carve((size_t)E * sizeof(float));

    init_zero<<<2048, 256, 0, stream>>>(out, denom, mkey, N * D, N);
    gemm_wmma<<<(N + TILE - 1) / TILE, 256, 0, stream>>>(h, W, z, N);
    node_scores<<<(N + 7) / 8, 256, 0, stream>>>(z, attn, ssrc, sdst, N);
    edge_logits<<<(E + 255) / 256, 256, 0, stream>>>(ssrc, sdst, src, dst, ebuf, mkey, E);
    edge_exp<<<(E + 255) / 256, 256, 0, stream>>>(ebuf, dst, mkey, denom, E);
    edge_scatter<<<(E + 7) / 8, 256, 0, stream>>>(z, ebuf, denom, rel, src, dst, ety, out, E);
}